// CGNet_flatten_GRU_39754217291887
// MI455X (gfx1250) — compile-verified
//
#include <hip/hip_runtime.h>
#include <hip/hip_bf16.h>
#include <math.h>

typedef __attribute__((ext_vector_type(2))) float v2f;
typedef __attribute__((ext_vector_type(8))) float v8f;

#define N_NODES 40000
#define N_EDGES 400000
#define CDIM    64
#define ZK      132              // K = 131 real + 1 bias row (multiple of 4)
#define NVALS   (N_NODES * CDIM)
#define WB_ELTS (66 * 128 * 2)   // frag-packed conv weight (132 rows x 128 cols)
#define WG_ELTS (32 * 192 * 2)   // frag-packed GRU weight  (64 rows x 192 cols)

// ---------------- utility kernels ----------------

__global__ void k_zero(float* __restrict__ p, int n) {
  int i = blockIdx.x * blockDim.x + threadIdx.x;
  if (i < n) p[i] = 0.f;
}

__global__ void k_count(const int* __restrict__ rows, float* __restrict__ cnt, int E) {
  int e = blockIdx.x * blockDim.x + threadIdx.x;
  if (e < E) atomicAdd(&cnt[rows[e]], 1.0f);
}

// Frag-pack conv weight: Wb[(k2*128 + c)*2 + j] = W[2*k2 + j][c]
// where W = [wf | ws] (131 rows) and row 131 = [bf | bs] (bias folded into GEMM).
__global__ void k_pack(const float* __restrict__ wf, const float* __restrict__ ws,
                       const float* __restrict__ bf, const float* __restrict__ bs,
                       float* __restrict__ Wb) {
  int i = blockIdx.x * blockDim.x + threadIdx.x;
  if (i >= WB_ELTS) return;
  int j  = i & 1;
  int c  = (i >> 1) & 127;
  int k2 = i >> 8;
  int r  = 2 * k2 + j;
  float v = 0.f;
  if (r < 131)       v = (c < 64) ? wf[r * 64 + c] : ws[r * 64 + (c - 64)];
  else if (r == 131) v = (c < 64) ? bf[c] : bs[c - 64];
  Wb[i] = v;
}

// Frag-pack GRU weight [64,192]: Wb[(k2*192 + c)*2 + j] = w[(2*k2 + j)*192 + c]
__global__ void k_packgru(const float* __restrict__ w, float* __restrict__ Wb) {
  int i = blockIdx.x * blockDim.x + threadIdx.x;
  if (i >= WG_ELTS) return;
  int j  = i & 1;
  int r2 = i >> 1;
  int c  = r2 % 192;
  int k2 = r2 / 192;
  Wb[i] = w[(2 * k2 + j) * 192 + c];
}

// x = relu(x0 @ fc1_w + fc1_b)
__global__ void k_fc1(const float* __restrict__ xin, const float* __restrict__ w,
                      const float* __restrict__ b, float* __restrict__ out) {
  int i = blockIdx.x * blockDim.x + threadIdx.x;
  if (i >= NVALS) return;
  int node = i >> 6, c = i & 63;
  float acc = b[c];
#pragma unroll
  for (int j = 0; j < 9; ++j) acc += xin[node * 9 + j] * w[j * 64 + c];
  out[i] = fmaxf(acc, 0.f);
}

// out = relu(xin + agg / max(cnt,1))
__global__ void k_update(const float* __restrict__ xin, const float* __restrict__ agg,
                         const float* __restrict__ cnt, float* __restrict__ out) {
  int i = blockIdx.x * blockDim.x + threadIdx.x;
  if (i >= NVALS) return;
  int node = i >> 6;
  float c = fmaxf(cnt[node], 1.0f);
  out[i] = fmaxf(xin[i] + agg[i] / c, 0.f);
}

// ---------------- CGConv edge kernel (f32 WMMA) ----------------
// 128 threads = 4 waves; 64 edges per workgroup, 16 edges per wave.
// msg = sigmoid(z@wf+bf) * softplus(z@ws+bs), atomically accumulated at row.
// Bias rides in K-row 131 (z[131] = 1.0).
__global__ __launch_bounds__(128) void k_cgconv(
    const float* __restrict__ xin, const int* __restrict__ ei, const float* __restrict__ ew,
    const float* __restrict__ Wb, float* __restrict__ agg)
{
  __shared__ float zs[64 * ZK];
  __shared__ int rowids[64];
  __shared__ int colids[64];
  int tid = threadIdx.x;
  int ebase = blockIdx.x * 64;

  if (tid < 64) rowids[tid] = ei[ebase + tid];
  else          colids[tid - 64] = ei[N_EDGES + ebase + (tid - 64)];
  __syncthreads();

  // gather z tiles: z[e] = [x[row] (64) | x[col] (64) | w_e (3) | 1.0]
  for (int i = tid; i < 64 * 32; i += 128) {
    int e = i >> 5, p = i & 31;
    int node = (p < 16) ? rowids[e] : colids[e];
    int j = p & 15;
    float4 v = ((const float4*)(xin + (size_t)node * 64))[j];
    *(float4*)(zs + e * ZK + ((p < 16) ? 0 : 64) + j * 4) = v;
  }
  if (tid < 64) {
    int e = ebase + tid;
    zs[tid * ZK + 128] = ew[e * 3 + 0];
    zs[tid * ZK + 129] = ew[e * 3 + 1];
    zs[tid * ZK + 130] = ew[e * 3 + 2];
    zs[tid * ZK + 131] = 1.0f;            // bias row
  }
  __syncthreads();

  int lane = tid & 31;
  int wave = tid >> 5;
  int me = lane & 15;     // M (edge in tile)
  int hi = lane >> 4;     // K half select per 16x4 f32 A layout
  int tb = wave * 16;     // edge tile base within block

  // A fragments for all 33 K-steps, kept in registers
  v2f a[33];
  const float* zrow = zs + (tb + me) * ZK + 2 * hi;
#pragma unroll
  for (int k = 0; k < 33; ++k) { a[k].x = zrow[4 * k]; a[k].y = zrow[4 * k + 1]; }

  const v2f* Wb2 = (const v2f*)Wb;        // index: k2*128 + col
  for (int t = 0; t < 4; ++t) {           // 4 channel tiles of 16 (per half)
    v8f cf = {}; v8f cs = {};
    int colf = t * 16 + me;               // wf half: cols 0..63
    int cols = 64 + t * 16 + me;          // ws half: cols 64..127
#pragma unroll
    for (int k = 0; k < 33; ++k) {
      int k2 = 2 * k + hi;
      v2f bfr = Wb2[k2 * 128 + colf];
      v2f bsr = Wb2[k2 * 128 + cols];
      cf = __builtin_amdgcn_wmma_f32_16x16x4_f32(false, a[k], false, bfr, (short)0, cf, false, false);
      cs = __builtin_amdgcn_wmma_f32_16x16x4_f32(false, a[k], false, bsr, (short)0, cs, false, false);
    }
    int ch = t * 16 + me;
#pragma unroll
    for (int r8 = 0; r8 < 8; ++r8) {
      int m = tb + r8 + 8 * hi;           // edge within block (C-tile layout)
      float f = cf[r8];
      float s = cs[r8];
      float sig = 1.0f / (1.0f + expf(-f));
      float sp  = (s > 20.f) ? s : log1pf(expf(s));
      atomicAdd(&agg[(size_t)rowids[m] * 64 + ch], sig * sp);
    }
  }
}

// ---------------- GRU kernel (f32 WMMA) ----------------
// h' = (1-z)*tanh(in + r*hn) + z*h ; 64 nodes per workgroup, 16 per wave.
__global__ __launch_bounds__(128) void k_gru(
    const float* __restrict__ m, float* __restrict__ h,
    const float* __restrict__ Wgi, const float* __restrict__ bih,
    const float* __restrict__ Wgh, const float* __restrict__ bhh)
{
  __shared__ float ms[64 * 68];
  __shared__ float hs[64 * 68];
  int tid = threadIdx.x;
  int nbase = blockIdx.x * 64;

  for (int i = tid; i < 64 * 16; i += 128) {
    int n = i >> 4, j = i & 15;
    float4 vm = ((const float4*)(m + (size_t)(nbase + n) * 64))[j];
    float4 vh = ((const float4*)(h + (size_t)(nbase + n) * 64))[j];
    float* dm = ms + n * 68 + 4 * j;
    dm[0] = vm.x; dm[1] = vm.y; dm[2] = vm.z; dm[3] = vm.w;
    float* dh = hs + n * 68 + 4 * j;
    dh[0] = vh.x; dh[1] = vh.y; dh[2] = vh.z; dh[3] = vh.w;
  }
  __syncthreads();

  int lane = tid & 31, wave = tid >> 5;
  int me = lane & 15, hi = lane >> 4, tb = wave * 16;

  v2f am[16], ah[16];
  const float* mr = ms + (tb + me) * 68 + 2 * hi;
  const float* hr = hs + (tb + me) * 68 + 2 * hi;
#pragma unroll
  for (int k = 0; k < 16; ++k) {
    am[k].x = mr[4 * k]; am[k].y = mr[4 * k + 1];
    ah[k].x = hr[4 * k]; ah[k].y = hr[4 * k + 1];
  }

  const v2f* Wi2 = (const v2f*)Wgi;       // index: k2*192 + col
  const v2f* Wh2 = (const v2f*)Wgh;
  for (int t = 0; t < 4; ++t) {
    v8f cir = {}, chr_ = {}, ciz = {}, chz = {}, cin = {}, chn = {};
    int cr = t * 16 + me, cz = 64 + t * 16 + me, cn = 128 + t * 16 + me;
#pragma unroll
    for (int k = 0; k < 16; ++k) {
      int k2 = 2 * k + hi;
      v2f b0 = Wi2[k2 * 192 + cr];
      v2f b1 = Wh2[k2 * 192 + cr];
      v2f b2 = Wi2[k2 * 192 + cz];
      v2f b3 = Wh2[k2 * 192 + cz];
      v2f b4 = Wi2[k2 * 192 + cn];
      v2f b5 = Wh2[k2 * 192 + cn];
      cir  = __builtin_amdgcn_wmma_f32_16x16x4_f32(false, am[k], false, b0, (short)0, cir,  false, false);
      chr_ = __builtin_amdgcn_wmma_f32_16x16x4_f32(false, ah[k], false, b1, (short)0, chr_, false, false);
      ciz  = __builtin_amdgcn_wmma_f32_16x16x4_f32(false, am[k], false, b2, (short)0, ciz,  false, false);
      chz  = __builtin_amdgcn_wmma_f32_16x16x4_f32(false, ah[k], false, b3, (short)0, chz,  false, false);
      cin  = __builtin_amdgcn_wmma_f32_16x16x4_f32(false, am[k], false, b4, (short)0, cin,  false, false);
      chn  = __builtin_amdgcn_wmma_f32_16x16x4_f32(false, ah[k], false, b5, (short)0, chn,  false, false);
    }
    int ch = t * 16 + me;
    float bir = bih[ch],       bhr = bhh[ch];
    float biz = bih[64 + ch],  bhz = bhh[64 + ch];
    float bin = bih[128 + ch], bhn = bhh[128 + ch];
#pragma unroll
    for (int r8 = 0; r8 < 8; ++r8) {
      int nl = tb + r8 + 8 * hi;
      float rg = 1.f / (1.f + expf(-(cir[r8] + bir + chr_[r8] + bhr)));
      float zg = 1.f / (1.f + expf(-(ciz[r8] + biz + chz[r8] + bhz)));
      float ng = tanhf(cin[r8] + bin + rg * (chn[r8] + bhn));
      float hv = hs[nl * 68 + ch];
      h[(size_t)(nbase + nl) * 64 + ch] = (1.f - zg) * ng + zg * hv;
    }
  }
}

// ---------------- readout (64 graphs, tiny) ----------------
__global__ __launch_bounds__(64) void k_readout(
    const float* __restrict__ xh,
    const float* __restrict__ fcs_w, const float* __restrict__ fcs_b,
    const float* __restrict__ f2c_w, const float* __restrict__ f2c_b,
    const float* __restrict__ f3c_w, const float* __restrict__ f3c_b,
    const float* __restrict__ f2d_w, const float* __restrict__ f2d_b,
    const float* __restrict__ f3d_w, const float* __restrict__ f3d_b,
    float* __restrict__ out)
{
  __shared__ float sA[64][65];
  __shared__ float sB[64][65];
  int g = threadIdx.x;
  const float* xg = xh + (size_t)g * (N_NODES / 64) * 64;  // first node of graph g
  for (int i = 0; i < 64; ++i) sA[g][i] = xg[i];
  for (int o = 0; o < 64; ++o) {
    float acc = fcs_b[o];
    for (int i = 0; i < 64; ++i) acc += sA[g][i] * fcs_w[i * 64 + o];
    sB[g][o] = fmaxf(acc, 0.f);
  }
  for (int o = 0; o < 64; ++o) {
    float acc = f2c_b[o];
    for (int i = 0; i < 64; ++i) acc += sB[g][i] * f2c_w[i * 64 + o];
    sA[g][o] = fmaxf(acc, 0.f);
  }
  for (int o = 0; o < 2; ++o) {
    float acc = f3c_b[o];
    for (int i = 0; i < 64; ++i) acc += sA[g][i] * f3c_w[i * 2 + o];
    out[g * 2 + o] = 1.f / (1.f + expf(-acc));
  }
  for (int o = 0; o < 64; ++o) {
    float acc = f2d_b[o];
    for (int i = 0; i < 64; ++i) acc += sB[g][i] * f2d_w[i * 64 + o];
    sA[g][o] = fmaxf(acc, 0.f);
  }
  for (int o = 0; o < 2; ++o) {
    float acc = f3d_b[o];
    for (int i = 0; i < 64; ++i) acc += sA[g][i] * f3d_w[i * 2 + o];
    out[128 + g * 2 + o] = acc;
  }
}

// ---------------- launcher ----------------
extern "C" void kernel_launch(void* const* d_in, const int* in_sizes, int n_in,
                              void* d_out, int out_size, void* d_ws, size_t ws_size,
                              hipStream_t stream) {
  (void)in_sizes; (void)n_in; (void)out_size; (void)ws_size;
  const float* x0   = (const float*)d_in[0];
  const int*   ei1  = (const int*)d_in[1];
  const int*   ei2  = (const int*)d_in[2];
  const float* w1   = (const float*)d_in[3];
  const float* w2   = (const float*)d_in[4];
  const float* fc1w = (const float*)d_in[6];
  const float* fc1b = (const float*)d_in[7];
  const float* lfw  = (const float*)d_in[8];
  const float* lfb  = (const float*)d_in[9];
  const float* lsw  = (const float*)d_in[10];
  const float* lsb  = (const float*)d_in[11];
  const float* wih  = (const float*)d_in[12];
  const float* bih  = (const float*)d_in[13];
  const float* whh  = (const float*)d_in[14];
  const float* bhh  = (const float*)d_in[15];
  const float* fcsw = (const float*)d_in[16];
  const float* fcsb = (const float*)d_in[17];
  const float* f2cw = (const float*)d_in[18];
  const float* f2cb = (const float*)d_in[19];
  const float* f3cw = (const float*)d_in[20];
  const float* f3cb = (const float*)d_in[21];
  const float* f2dw = (const float*)d_in[22];
  const float* f2db = (const float*)d_in[23];
  const float* f3dw = (const float*)d_in[24];
  const float* f3db = (const float*)d_in[25];
  float* out = (float*)d_out;

  float* xh   = (float*)d_ws;                 // node state (x == h)
  float* mbuf = xh + (size_t)NVALS;           // conv output
  float* agg  = mbuf + (size_t)NVALS;         // scatter accumulator
  float* cnt1 = agg + (size_t)NVALS;
  float* cnt2 = cnt1 + N_NODES;
  float* Wb   = cnt2 + N_NODES;               // frag-packed conv weight (+bias row)
  float* Wgi  = Wb + WB_ELTS;                 // frag-packed gru w_ih
  float* Wgh  = Wgi + WG_ELTS;                // frag-packed gru w_hh

  dim3 b256(256);
  k_zero<<<(N_NODES + 255) / 256, b256, 0, stream>>>(cnt1, N_NODES);
  k_zero<<<(N_NODES + 255) / 256, b256, 0, stream>>>(cnt2, N_NODES);
  k_count<<<(N_EDGES + 255) / 256, b256, 0, stream>>>(ei1, cnt1, N_EDGES);
  k_count<<<(N_EDGES + 255) / 256, b256, 0, stream>>>(ei2, cnt2, N_EDGES);
  k_pack<<<(WB_ELTS + 255) / 256, b256, 0, stream>>>(lfw, lsw, lfb, lsb, Wb);
  k_packgru<<<(WG_ELTS + 255) / 256, b256, 0, stream>>>(wih, Wgi);
  k_packgru<<<(WG_ELTS + 255) / 256, b256, 0, stream>>>(whh, Wgh);
  k_fc1<<<(NVALS + 255) / 256, b256, 0, stream>>>(x0, fc1w, fc1b, xh);

  for (int it = 0; it < 3; ++it) {
    k_zero<<<(NVALS + 255) / 256, b256, 0, stream>>>(agg, NVALS);
    k_cgconv<<<N_EDGES / 64, 128, 0, stream>>>(xh, ei2, w2, Wb, agg);
    k_update<<<(NVALS + 255) / 256, b256, 0, stream>>>(xh, agg, cnt2, mbuf);
    k_zero<<<(NVALS + 255) / 256, b256, 0, stream>>>(agg, NVALS);
    k_cgconv<<<N_EDGES / 64, 128, 0, stream>>>(mbuf, ei1, w1, Wb, agg);
    k_update<<<(NVALS + 255) / 256, b256, 0, stream>>>(mbuf, agg, cnt1, mbuf);
    k_gru<<<N_NODES / 64, 128, 0, stream>>>(mbuf, xh, Wgi, bih, Wgh, bhh);
  }
  k_readout<<<1, 64, 0, stream>>>(xh, fcsw, fcsb, f2cw, f2cb, f3cw, f3cb,
                                  f2dw, f2db, f3dw, f3db, out);
}